// PSGIN_85856396248050
// MI455X (gfx1250) — compile-verified
//
#include <hip/hip_runtime.h>
#include <math.h>

// Problem constants (from reference)
constexpr int B_ = 8, T_ = 24, N_ = 207, F_ = 16, H_ = 64;
constexpr int DH_ = 16, DA_ = 64, DS_ = 64;
constexpr int NP = 208;                       // node dim padded to 13*16
constexpr long NP2 = (long)NP * NP;
constexpr float ALPHA_ = 0.05f, BETA_ = 0.95f, GAMMA_ = 0.95f;

typedef __attribute__((ext_vector_type(2))) float v2f;
typedef __attribute__((ext_vector_type(8))) float v8f;

// Epilogue + store of one 16x16 tile (C/D layout: VGPR r -> row tm+r+8*(lane/16))
__device__ __forceinline__ void psgin_store_tile(
    v8f acc, int tm, int tn, int l16, int lhi,
    const float* __restrict__ R1, long r1off, float r1c,
    const float* __restrict__ bias,
    float* __restrict__ D, int ldd,
    int Ms, int Ns, float alpha, int act)
{
#pragma unroll
  for (int rI = 0; rI < 8; ++rI) {
    const int row = tm + rI + 8 * lhi;
    const int col = tn + l16;
    if (row < Ms && col < Ns) {
      float v = alpha * acc[rI];
      if (R1)   v += r1c * R1[r1off + (long)row * ldd + col];
      if (bias) v += bias[col];
      if (act == 1)      v = 1.0f / (1.0f + __expf(-v));
      else if (act == 2) v = tanhf(v);
      else if (act == 3) v = fmaxf(v, 0.0f);
      D[(long)row * ldd + col] = v;
    }
  }
}

// ---------------------------------------------------------------------------
// Batched WMMA GEMM, branch-free inner loop (operands padded by construction):
//   D[bz] = act( alpha * A[bz] @ op(B[bz]) + r1c * R1[bz] + bias )
// A row-major (M x K). TB=1: B is (N x K) row-major (B^T used). Kp % 16 == 0.
// Each wave computes a 16x32 strip: two independent accumulator chains share
// the A fragment (halves A traffic, overlaps XDL dependency latency).
// act: 0=none 1=sigmoid 2=tanh 3=relu
// ---------------------------------------------------------------------------
template <int TB>
__global__ void psgin_wmma_gemm(
    const float* __restrict__ A, int lda, long sA,
    const float* __restrict__ Bm, int ldb, long sB,
    const float* __restrict__ R1, long sR1, float r1c,
    const float* __restrict__ bias,
    float* __restrict__ D, int ldd, long sD,
    int Ms, int Ns, int Kp, float alpha, int act,
    int tilesN, int pairs, int tilesTot)
{
  const int bz = blockIdx.z;
  A  += (long)bz * sA;
  Bm += (long)bz * sB;
  D  += (long)bz * sD;
  const long r1off = (long)bz * sR1;

  const int lane = threadIdx.x & 31;
  const int wave = threadIdx.x >> 5;
  const int pid = blockIdx.x * 4 + wave;
  if (pid >= tilesTot) return;             // uniform per wave
  const int tm  = (pid / pairs) * 16;
  const int tn0 = (pid % pairs) * 32;
  const bool has2 = (tn0 + 16) < tilesN * 16;   // wave-uniform
  const int l16 = lane & 15;
  const int lhi = lane >> 4;

  // A frag: (m = tm+l16, k = k0 + 2*lhi + {0,1}) -> contiguous pair
  const float* ap = A + (long)(tm + l16) * lda + 2 * lhi;
  // B frag: (k = k0 + 2*lhi + {0,1}, n = tn+l16)
  const float* bp0 = TB ? (Bm + (long)(tn0 + l16) * ldb + 2 * lhi)
                        : (Bm + (tn0 + l16) + (long)(2 * lhi) * ldb);

  v8f acc0 = {0.f,0.f,0.f,0.f,0.f,0.f,0.f,0.f};
  v8f acc1 = {0.f,0.f,0.f,0.f,0.f,0.f,0.f,0.f};

  if (has2) {
    const float* bp1 = TB ? (bp0 + (long)16 * ldb) : (bp0 + 16);
    for (int k0 = 0; k0 < Kp; k0 += 16) {
#pragma unroll
      for (int u = 0; u < 4; ++u) {
        const int ka = k0 + 4 * u;
        v2f av = *(const v2f*)(ap + ka);
        v2f bv0, bv1;
        if (TB) {
          bv0 = *(const v2f*)(bp0 + ka);
          bv1 = *(const v2f*)(bp1 + ka);
        } else {
          bv0[0] = bp0[(long)ka * ldb]; bv0[1] = bp0[(long)(ka + 1) * ldb];
          bv1[0] = bp1[(long)ka * ldb]; bv1[1] = bp1[(long)(ka + 1) * ldb];
        }
        acc0 = __builtin_amdgcn_wmma_f32_16x16x4_f32(
            false, av, false, bv0, (short)0, acc0, false, false);
        acc1 = __builtin_amdgcn_wmma_f32_16x16x4_f32(
            false, av, false, bv1, (short)0, acc1, false, false);
      }
    }
  } else {
    for (int k0 = 0; k0 < Kp; k0 += 16) {
#pragma unroll
      for (int u = 0; u < 4; ++u) {
        const int ka = k0 + 4 * u;
        v2f av = *(const v2f*)(ap + ka);
        v2f bv0;
        if (TB) {
          bv0 = *(const v2f*)(bp0 + ka);
        } else {
          bv0[0] = bp0[(long)ka * ldb]; bv0[1] = bp0[(long)(ka + 1) * ldb];
        }
        acc0 = __builtin_amdgcn_wmma_f32_16x16x4_f32(
            false, av, false, bv0, (short)0, acc0, false, false);
      }
    }
  }

  psgin_store_tile(acc0, tm, tn0, l16, lhi, R1, r1off, r1c, bias, D, ldd, Ms, Ns, alpha, act);
  if (has2)
    psgin_store_tile(acc1, tm, tn0 + 16, l16, lhi, R1, r1off, r1c, bias, D, ldd, Ms, Ns, alpha, act);
}

// ------------------------------- pointwise ---------------------------------
__global__ void psgin_fill(float* p, float v, long n) {
  long i = (long)blockIdx.x * 256 + threadIdx.x;
  if (i < n) p[i] = v;
}
__global__ void psgin_multanh(const float* a, const float* b, float* o, int n) {
  int i = blockIdx.x * 256 + threadIdx.x;
  if (i < n) o[i] = tanhf(2.0f * a[i] * b[i]);
}
__global__ void psgin_gru(const float* z, const float* c, float* h, int n) {
  int i = blockIdx.x * 256 + threadIdx.x;
  if (i < n) { float zz = z[i]; h[i] = zz * h[i] + (1.0f - zz) * c[i]; }
}
// pad-copy a (B, 207, 16) slice (batch stride xbs) into padded (B, 208, 16)
__global__ void psgin_padx(const float* __restrict__ x, long xbs,
                           float* __restrict__ xp, int n) {
  int idx = blockIdx.x * 256 + threadIdx.x;
  if (idx >= n) return;
  int cc = idx % F_, i = (idx / F_) % NP, b = idx / (NP * F_);
  xp[idx] = (i < N_) ? x[(long)b * xbs + (long)i * F_ + cc] : 0.0f;
}
// recent: inp[b, T-4+t] -> rcopy[(b*4+t)] padded rows
__global__ void psgin_recent(const float* __restrict__ in, float* __restrict__ out, int n) {
  int idx = blockIdx.x * 256 + threadIdx.x;
  if (idx >= n) return;
  int cc = idx % F_, i = (idx / F_) % NP, bt = idx / (NP * F_);
  int b = bt >> 2, t = bt & 3;
  out[idx] = (i < N_)
      ? in[((long)b * T_ + (T_ - 4 + t)) * N_ * F_ + (long)i * F_ + cc] : 0.0f;
}
// astaT[w*NP + v] = asta[v*207 + w], zero pads
__global__ void psgin_pad_asta(const float* __restrict__ a, float* __restrict__ at, int n) {
  int idx = blockIdx.x * 256 + threadIdx.x;
  if (idx >= n) return;
  int v = idx % NP, w = idx / NP;
  at[idx] = (v < N_ && w < N_) ? a[(long)v * N_ + w] : 0.0f;
}
// adjT[b, w, v] = normalized( relu(tanh(2*(G[v,w]-G[w,v]))) + (v==w) ), built
// transposed so propagation GEMMs read row-major A. Row w=207 written as 0.
__global__ void psgin_adj(const float* __restrict__ G, float* __restrict__ adjT) {
  int i = blockIdx.x, b = blockIdx.y, j = threadIdx.x;   // i = v (row), j = w
  const float* Gb = G + (long)b * NP2;
  float v = 0.0f;
  if (j < N_) {
    float g = Gb[(long)i * NP + j] - Gb[(long)j * NP + i];
    v = fmaxf(tanhf(2.0f * g), 0.0f) + (i == j ? 1.0f : 0.0f);
  }
  __shared__ float red[256];
  red[j] = v; __syncthreads();
  for (int s = 128; s > 0; s >>= 1) { if (j < s) red[j] += red[j + s]; __syncthreads(); }
  float inv = 1.0f / (red[0] + 1e-8f);
  if (j < NP) adjT[(long)b * NP2 + (long)j * NP + i] = (j < N_) ? v * inv : 0.0f;
}
// Fused attention: score[j] = sum_d tanh(q[b,i,d]+k[bt,j,d])*sw[d]
// dynT[bt, w=j, v=i] = static[i,j] + softmax_j(score)  (transposed write)
__global__ void psgin_attn(const float* __restrict__ q, const float* __restrict__ k,
                           const float* __restrict__ sw, const float* __restrict__ sadj,
                           float* __restrict__ dynT) {
  int i = blockIdx.x, bt = blockIdx.y, j = threadIdx.x;
  int b = bt >> 2;
  __shared__ float qs[DA_];
  __shared__ float sws[DA_];
  __shared__ float red[256];
  if (j < DA_) { qs[j] = q[((long)b * NP + i) * DA_ + j]; sws[j] = sw[j]; }
  __syncthreads();
  float sc = -1e30f;
  if (j < N_) {
    const float* kr = k + ((long)bt * NP + j) * DA_;
    float acc = 0.0f;
    for (int d = 0; d < DA_; ++d) acc += tanhf(qs[d] + kr[d]) * sws[d];
    sc = acc;
  }
  red[j] = sc; __syncthreads();
  for (int s = 128; s > 0; s >>= 1) { if (j < s) red[j] = fmaxf(red[j], red[j + s]); __syncthreads(); }
  float mx = red[0]; __syncthreads();
  float e = (j < N_) ? __expf(sc - mx) : 0.0f;
  red[j] = e; __syncthreads();
  for (int s = 128; s > 0; s >>= 1) { if (j < s) red[j] += red[j + s]; __syncthreads(); }
  float inv = 1.0f / red[0];
  if (j < NP)
    dynT[(long)bt * NP2 + (long)j * NP + i] =
        (j < N_) ? sadj[(long)i * N_ + j] + e * inv : 0.0f;
}
// o[(b*NP+i)*80 + c] = c<F ? xp : (r ? r*h : h)   (padded rows pass through)
__global__ void psgin_concat(const float* __restrict__ xp,
                             const float* __restrict__ h,
                             const float* __restrict__ r,
                             float* __restrict__ o, int n) {
  int idx = blockIdx.x * 256 + threadIdx.x;
  if (idx >= n) return;
  const int C = F_ + H_;
  int c = idx % C, bi = idx / C;
  int b = bi / NP, i = bi % NP;
  float v;
  if (c < F_) v = xp[((long)b * NP + i) * F_ + c];
  else {
    long ho = (long)bi * H_ + (c - F_);
    v = h[ho];
    if (r) v *= r[ho];
  }
  o[idx] = v;
}
// tg2[b,:] = sum_{t=0..3} sg[b*4+t,:]   (padded block = NP*DS floats)
__global__ void psgin_sum4(const float* __restrict__ in, float* __restrict__ out, int n) {
  int idx = blockIdx.x * 256 + threadIdx.x;
  if (idx >= n) return;
  const int S = NP * DS_;
  int b = idx / S, rem = idx % S;
  const float* p = in + (long)b * 4 * S + rem;
  out[idx] = p[0] + p[S] + p[2 * S] + p[3 * S];
}

// ------------------------------- host side ---------------------------------
static inline void gemm(hipStream_t st,
                        const float* A, int lda, long sA,
                        const float* B, int ldb, long sB, int tB,
                        const float* R1, long sR1, float r1c,
                        const float* bias,
                        float* D, int ldd, long sD,
                        int Ms, int Ns, int Kp, int batch, float alpha, int act) {
  int tilesN = (Ns + 15) >> 4;
  int tilesM = (Ms + 15) >> 4;
  int pairs = (tilesN + 1) >> 1;
  int tilesTot = tilesM * pairs;
  dim3 grid((tilesTot + 3) / 4, 1, batch);
  if (tB)
    psgin_wmma_gemm<1><<<grid, 128, 0, st>>>(A, lda, sA, B, ldb, sB, R1, sR1, r1c,
                                             bias, D, ldd, sD, Ms, Ns, Kp, alpha, act,
                                             tilesN, pairs, tilesTot);
  else
    psgin_wmma_gemm<0><<<grid, 128, 0, st>>>(A, lda, sA, B, ldb, sB, R1, sR1, r1c,
                                             bias, D, ldd, sD, Ms, Ns, Kp, alpha, act,
                                             tilesN, pairs, tilesTot);
}
static inline dim3 pwg(long n) { return dim3((unsigned)((n + 255) / 256)); }

struct Ctx {
  hipStream_t st;
  const float *astaT;
  const float *Wsrc, *bsrc, *Wtgt, *btgt, *Wse, *bse, *Wte, *bte;
  const float *Wz, *bz, *Wr, *br, *Wc, *bc;
  float *xp, *h, *h1p, *h2p, *sctx, *tctx, *sb, *tb, *G, *adjT;
  float *gi, *g1, *g2, *ri, *zb, *rb, *cb;
};

static void run_cell(const Ctx& c) {
  const long NPH = (long)NP * H_, NP16 = (long)NP * DH_, NP80 = (long)NP * 80;
  const long NPF = (long)NP * F_;
  hipStream_t st = c.st;
  // --- gcn_hyper shared propagation: h1 = G*(AsT@h) + A*h ; h2 = G*(AsT@h1) + A*h
  gemm(st, c.astaT, NP, 0, c.h,   H_, NPH, 0, c.h, NPH, ALPHA_, nullptr, c.h1p, H_, NPH, NP, H_, NP, B_, GAMMA_, 0);
  gemm(st, c.astaT, NP, 0, c.h1p, H_, NPH, 0, c.h, NPH, ALPHA_, nullptr, c.h2p, H_, NPH, NP, H_, NP, B_, GAMMA_, 0);
  // --- src/tgt context: [h|h1|h2] @ W(192,16) + b  as 3 accumulating GEMMs
  gemm(st, c.h,   H_, NPH, c.Wsrc,                DH_, 0, 0, nullptr, 0, 0.f,   nullptr, c.sctx, DH_, NP16, NP, DH_, H_, B_, 1.f, 0);
  gemm(st, c.h1p, H_, NPH, c.Wsrc + (long)H_*DH_,   DH_, 0, 0, c.sctx, NP16, 1.f, nullptr, c.sctx, DH_, NP16, NP, DH_, H_, B_, 1.f, 0);
  gemm(st, c.h2p, H_, NPH, c.Wsrc + (long)2*H_*DH_, DH_, 0, 0, c.sctx, NP16, 1.f, c.bsrc,  c.sctx, DH_, NP16, NP, DH_, H_, B_, 1.f, 0);
  gemm(st, c.h,   H_, NPH, c.Wtgt,                DH_, 0, 0, nullptr, 0, 0.f,   nullptr, c.tctx, DH_, NP16, NP, DH_, H_, B_, 1.f, 0);
  gemm(st, c.h1p, H_, NPH, c.Wtgt + (long)H_*DH_,   DH_, 0, 0, c.tctx, NP16, 1.f, nullptr, c.tctx, DH_, NP16, NP, DH_, H_, B_, 1.f, 0);
  gemm(st, c.h2p, H_, NPH, c.Wtgt + (long)2*H_*DH_, DH_, 0, 0, c.tctx, NP16, 1.f, c.btgt,  c.tctx, DH_, NP16, NP, DH_, H_, B_, 1.f, 0);
  // --- embeddings, then s = tanh(2*emb*ctx) in place; t likewise
  gemm(st, c.xp, F_, NPF, c.Wse, DH_, 0, 0, nullptr, 0, 0.f, c.bse, c.sb, DH_, NP16, NP, DH_, F_, B_, 1.f, 0);
  gemm(st, c.xp, F_, NPF, c.Wte, DH_, 0, 0, nullptr, 0, 0.f, c.bte, c.tb, DH_, NP16, NP, DH_, F_, B_, 1.f, 0);
  psgin_multanh<<<pwg(B_*NP16), 256, 0, st>>>(c.sb, c.sctx, c.sb, B_*NP16);
  psgin_multanh<<<pwg(B_*NP16), 256, 0, st>>>(c.tb, c.tctx, c.tb, B_*NP16);
  // --- G = s @ t^T (TB=1) ; adjacency built transposed + row-normalized
  gemm(st, c.sb, DH_, NP16, c.tb, DH_, NP16, 1, nullptr, 0, 0.f, nullptr, c.G, NP, NP2, NP, NP, DH_, B_, 1.f, 0);
  psgin_adj<<<dim3(N_, B_), 256, 0, st>>>(c.G, c.adjT);
  // --- gi = [x, h]; two-step propagation (shared between z and r gates)
  psgin_concat<<<pwg(B_*NP80), 256, 0, st>>>(c.xp, c.h, nullptr, c.gi, B_*NP80);
  gemm(st, c.adjT,  NP, NP2, c.gi, 80, NP80, 0, c.gi, NP80, ALPHA_, nullptr, c.g1, 80, NP80, NP, 80, NP, B_, BETA_,  0);
  gemm(st, c.astaT, NP, 0,   c.gi, 80, NP80, 0, c.g1, NP80, 1.f,    nullptr, c.g1, 80, NP80, NP, 80, NP, B_, GAMMA_, 0);
  gemm(st, c.adjT,  NP, NP2, c.g1, 80, NP80, 0, c.gi, NP80, ALPHA_, nullptr, c.g2, 80, NP80, NP, 80, NP, B_, BETA_,  0);
  gemm(st, c.astaT, NP, 0,   c.g1, 80, NP80, 0, c.g2, NP80, 1.f,    nullptr, c.g2, 80, NP80, NP, 80, NP, B_, GAMMA_, 0);
  // --- z, r gates: [gi|g1|g2] @ W(240,64) + b, sigmoid
  gemm(st, c.gi, 80, NP80, c.Wz,          H_, 0, 0, nullptr, 0, 0.f, nullptr, c.zb, H_, NPH, NP, H_, 80, B_, 1.f, 0);
  gemm(st, c.g1, 80, NP80, c.Wz + 80*H_,  H_, 0, 0, c.zb, NPH, 1.f,  nullptr, c.zb, H_, NPH, NP, H_, 80, B_, 1.f, 0);
  gemm(st, c.g2, 80, NP80, c.Wz + 160*H_, H_, 0, 0, c.zb, NPH, 1.f,  c.bz,    c.zb, H_, NPH, NP, H_, 80, B_, 1.f, 1);
  gemm(st, c.gi, 80, NP80, c.Wr,          H_, 0, 0, nullptr, 0, 0.f, nullptr, c.rb, H_, NPH, NP, H_, 80, B_, 1.f, 0);
  gemm(st, c.g1, 80, NP80, c.Wr + 80*H_,  H_, 0, 0, c.rb, NPH, 1.f,  nullptr, c.rb, H_, NPH, NP, H_, 80, B_, 1.f, 0);
  gemm(st, c.g2, 80, NP80, c.Wr + 160*H_, H_, 0, 0, c.rb, NPH, 1.f,  c.br,    c.rb, H_, NPH, NP, H_, 80, B_, 1.f, 1);
  // --- ri = [x, r*h]; propagate (reuse g1,g2); c = tanh([ri|g1|g2]@Wc + b)
  psgin_concat<<<pwg(B_*NP80), 256, 0, st>>>(c.xp, c.h, c.rb, c.ri, B_*NP80);
  gemm(st, c.adjT,  NP, NP2, c.ri, 80, NP80, 0, c.ri, NP80, ALPHA_, nullptr, c.g1, 80, NP80, NP, 80, NP, B_, BETA_,  0);
  gemm(st, c.astaT, NP, 0,   c.ri, 80, NP80, 0, c.g1, NP80, 1.f,    nullptr, c.g1, 80, NP80, NP, 80, NP, B_, GAMMA_, 0);
  gemm(st, c.adjT,  NP, NP2, c.g1, 80, NP80, 0, c.ri, NP80, ALPHA_, nullptr, c.g2, 80, NP80, NP, 80, NP, B_, BETA_,  0);
  gemm(st, c.astaT, NP, 0,   c.g1, 80, NP80, 0, c.g2, NP80, 1.f,    nullptr, c.g2, 80, NP80, NP, 80, NP, B_, GAMMA_, 0);
  gemm(st, c.ri, 80, NP80, c.Wc,          H_, 0, 0, nullptr, 0, 0.f, nullptr, c.cb, H_, NPH, NP, H_, 80, B_, 1.f, 0);
  gemm(st, c.g1, 80, NP80, c.Wc + 80*H_,  H_, 0, 0, c.cb, NPH, 1.f,  nullptr, c.cb, H_, NPH, NP, H_, 80, B_, 1.f, 0);
  gemm(st, c.g2, 80, NP80, c.Wc + 160*H_, H_, 0, 0, c.cb, NPH, 1.f,  c.bc,    c.cb, H_, NPH, NP, H_, 80, B_, 1.f, 2);
  // --- h = z*h + (1-z)*c
  psgin_gru<<<pwg(B_*NPH), 256, 0, st>>>(c.zb, c.cb, c.h, B_*NPH);
}

extern "C" void kernel_launch(void* const* d_in, const int* in_sizes, int n_in,
                              void* d_out, int out_size, void* d_ws, size_t ws_size,
                              hipStream_t stream) {
  const float* inp  = (const float*)d_in[0];
  const float* asta = (const float*)d_in[1];
  const float* Wsrc = (const float*)d_in[2];  const float* bsrc = (const float*)d_in[3];
  const float* Wtgt = (const float*)d_in[4];  const float* btgt = (const float*)d_in[5];
  const float* Wse  = (const float*)d_in[6];  const float* bse  = (const float*)d_in[7];
  const float* Wte  = (const float*)d_in[8];  const float* bte  = (const float*)d_in[9];
  const float* Wz   = (const float*)d_in[10]; const float* bz   = (const float*)d_in[11];
  const float* Wr   = (const float*)d_in[12]; const float* br   = (const float*)d_in[13];
  const float* Wc   = (const float*)d_in[14]; const float* bc   = (const float*)d_in[15];
  const float* Wq   = (const float*)d_in[16];
  const float* Wk   = (const float*)d_in[17]; const float* bk   = (const float*)d_in[18];
  const float* sw   = (const float*)d_in[19];
  const float* Wsg  = (const float*)d_in[20]; const float* bsg  = (const float*)d_in[21];
  const float* Wsh  = (const float*)d_in[22]; const float* bsh  = (const float*)d_in[23];
  const float* Wlo  = (const float*)d_in[24]; const float* blo  = (const float*)d_in[25];
  const float* Wfu  = (const float*)d_in[26]; const float* bfu  = (const float*)d_in[27];

  float* ws = (float*)d_ws;
  size_t off = 0;
  auto A = [&](size_t n) { float* p = ws + off; off += n; return p; };
  const long NPH = (long)NP * H_, NPF = (long)NP * F_;
  const long N16r = (long)N_ * F_;           // real (unpadded) node*F

  Ctx c;
  c.st = stream;
  c.Wsrc = Wsrc; c.bsrc = bsrc; c.Wtgt = Wtgt; c.btgt = btgt;
  c.Wse = Wse; c.bse = bse; c.Wte = Wte; c.bte = bte;
  c.Wz = Wz; c.bz = bz; c.Wr = Wr; c.br = br; c.Wc = Wc; c.bc = bc;

  float* astaT = A(NP2);            c.astaT = astaT;
  c.xp   = A(B_ * NPF);
  c.h    = A(B_ * NPH);
  c.h1p  = A(B_ * NPH);
  c.h2p  = A(B_ * NPH);
  c.sctx = A(B_ * NP * DH_);
  c.tctx = A(B_ * NP * DH_);
  c.sb   = A(B_ * NP * DH_);
  c.tb   = A(B_ * NP * DH_);
  c.G    = A(B_ * NP2);
  c.adjT = A(B_ * NP2);
  c.gi   = A(B_ * NP * (F_ + H_));
  c.g1   = A(B_ * NP * (F_ + H_));
  c.g2   = A(B_ * NP * (F_ + H_));
  c.ri   = A(B_ * NP * (F_ + H_));
  c.zb   = A(B_ * NPH);
  c.rb   = A(B_ * NPH);
  c.cb   = A(B_ * NPH);
  float* rcopy = A(32 * NPF);
  float* qb    = A(B_ * NP * DA_);
  float* kb    = A(32 * NP * DA_);
  float* dynT  = A(32 * NP2);
  float* p1    = A(32 * NPF);
  float* p2    = A(32 * NPF);
  float* sg    = A(32 * NP * DS_);
  float* tg2   = A(B_ * NP * DS_);

  // deterministic init: transposed+padded static adjacency; zero the pads of
  // the transposed adjacency buffers (their pad COLUMNS feed K-sums); h = 0
  psgin_pad_asta<<<pwg(NP2), 256, 0, stream>>>(asta, astaT, (int)NP2);
  psgin_fill<<<pwg(B_ * NP2), 256, 0, stream>>>(c.adjT, 0.0f, B_ * NP2);
  psgin_fill<<<pwg(32 * NP2), 256, 0, stream>>>(dynT, 0.0f, 32 * NP2);
  psgin_fill<<<pwg(B_ * NPH), 256, 0, stream>>>(c.h, 0.0f, B_ * NPH);

  // recurrent cells over t = 0,4,8,12,16
  for (int t = 0; t < T_ - 4; t += 4) {
    psgin_padx<<<pwg(B_ * NPF), 256, 0, stream>>>(inp + (long)t * N16r,
                                                  (long)T_ * N16r, c.xp, (int)(B_ * NPF));
    run_cell(c);
  }

  // ---- attention block ----
  psgin_padx<<<pwg(B_ * NPF), 256, 0, stream>>>(inp + (long)(T_ - 1) * N16r,
                                                (long)T_ * N16r, c.xp, (int)(B_ * NPF));
  gemm(stream, c.xp, F_, NPF, Wq, DA_, 0, 0, nullptr, 0, 0.f, nullptr,
       qb, DA_, (long)NP * DA_, NP, DA_, F_, B_, 1.f, 0);
  psgin_recent<<<pwg(32 * NPF), 256, 0, stream>>>(inp, rcopy, (int)(32 * NPF));
  gemm(stream, rcopy, F_, NPF, Wk, DA_, 0, 0, nullptr, 0, 0.f, bk,
       kb, DA_, (long)NP * DA_, NP, DA_, F_, 32, 1.f, 0);
  psgin_attn<<<dim3(N_, 32), 256, 0, stream>>>(qb, kb, sw, asta, dynT);

  // gcn_ssgal on recent with dynamic adjacency (C = F = 16)
  gemm(stream, dynT, NP, NP2, rcopy, F_, NPF, 0, rcopy, NPF, ALPHA_, nullptr,
       p1, F_, NPF, NP, F_, NP, 32, GAMMA_, 0);
  gemm(stream, dynT, NP, NP2, p1,    F_, NPF, 0, rcopy, NPF, ALPHA_, nullptr,
       p2, F_, NPF, NP, F_, NP, 32, GAMMA_, 0);
  gemm(stream, rcopy, F_, NPF, Wsg,            DS_, 0, 0, nullptr, 0, 0.f, nullptr,
       sg, DS_, (long)NP*DS_, NP, DS_, F_, 32, 1.f, 0);
  gemm(stream, p1,    F_, NPF, Wsg + F_*DS_,   DS_, 0, 0, sg, (long)NP*DS_, 1.f, nullptr,
       sg, DS_, (long)NP*DS_, NP, DS_, F_, 32, 1.f, 0);
  gemm(stream, p2,    F_, NPF, Wsg + 2*F_*DS_, DS_, 0, 0, sg, (long)NP*DS_, 1.f, bsg,
       sg, DS_, (long)NP*DS_, NP, DS_, F_, 32, 1.f, 3);
  psgin_sum4<<<pwg((long)B_ * NP * DS_), 256, 0, stream>>>(sg, tg2, B_ * NP * DS_);

  // outputs (stores into unpadded d_out are edge-guarded by the GEMM)
  float* short_o = (float*)d_out;
  float* long_o  = short_o + B_ * N16r;
  float* fus_o   = long_o + B_ * N16r;
  gemm(stream, tg2, DS_, (long)NP * DS_, Wsh, F_, 0, 0, nullptr, 0, 0.f, bsh,
       short_o, F_, N16r, N_, F_, DS_, B_, 1.f, 0);

  // final cell on `short`
  psgin_padx<<<pwg(B_ * NPF), 256, 0, stream>>>(short_o, N16r, c.xp, (int)(B_ * NPF));
  run_cell(c);

  gemm(stream, c.h, H_, NPH, Wlo, F_, 0, 0, nullptr, 0, 0.f, blo,
       long_o, F_, N16r, N_, F_, H_, B_, 1.f, 0);
  gemm(stream, tg2, DS_, (long)NP * DS_, Wfu, F_, 0, 0, nullptr, 0, 0.f, nullptr,
       fus_o, F_, N16r, N_, F_, DS_, B_, 1.f, 0);
  gemm(stream, c.h, H_, NPH, Wfu + (long)DS_ * F_, F_, 0, 0, fus_o, N16r, 1.f, bfu,
       fus_o, F_, N16r, N_, F_, H_, B_, 1.f, 0);
  (void)in_sizes; (void)n_in; (void)out_size; (void)ws_size;
}